// OneHopSumNodeLabelAggregator_81252191305759
// MI455X (gfx1250) — compile-verified
//
#include <hip/hip_runtime.h>

#define TPB 256

// Kernel 1: for rows [0, batch):
//   out[row, 0:48]  = x[row, 0:48]   (staged through LDS with gfx1250 async ops)
//   out[row, 48:96] = 0              (b128 stores; accumulator must be zeroed every call)
// One thread per float4 chunk of the x-half: total = batch * 12.
__global__ void copy_zero_async_kernel(const float4* __restrict__ x4,
                                       float4* __restrict__ out4,
                                       int total) {
    __shared__ float4 stage[TPB];
    int g = blockIdx.x * TPB + threadIdx.x;
    if (g >= total) return;
    int row = g / 12;
    int c   = g - row * 12;

    // zero the neighbor-sum half (columns 48..95), one float4 per thread
    out4[(size_t)row * 24 + 12 + c] = make_float4(0.f, 0.f, 0.f, 0.f);

    // async copy x[row, c*4 .. c*4+3] -> LDS -> out[row, c*4 .. c*4+3]
    const float4* srcp = x4 + (size_t)row * 12 + c;
    float4*       dstp = out4 + (size_t)row * 24 + c;
    // LDS byte offset of this lane's private staging slot (flat LDS addr
    // truncates to the in-allocation byte offset; HW adds LDS_BASE).
    unsigned lds_off = (unsigned)(unsigned long long)(void*)&stage[threadIdx.x];

    asm volatile("global_load_async_to_lds_b128 %0, %1, off"
                 :: "v"(lds_off), "v"(srcp) : "memory");
    asm volatile("s_wait_asynccnt 0" ::: "memory");
    asm volatile("global_store_async_from_lds_b128 %0, %1, off"
                 :: "v"(dstp), "v"(lds_off) : "memory");
}

// Kernel 2: scatter-add. Thread = (edge, 4-feature chunk); 12 threads/edge.
// The 12 sibling lanes issue identical edge-index addresses (merged) and read
// one contiguous 192B row of x (coalesced L2 gather). Atomic adds are
// explicit no-return global_atomic_add_f32 (fire-and-forget into the L2
// atomic units, tracked only by STOREcnt) — one address pair + offset
// immediates for the 4 consecutive floats.
__global__ void edge_scatter_kernel(const float4* __restrict__ x4,
                                    const int* __restrict__ eidx,
                                    float* __restrict__ out,
                                    int n_edges, int batch) {
    unsigned g = blockIdx.x * TPB + threadIdx.x;
    unsigned e = g / 12u;
    unsigned c = g - e * 12u;
    if (e >= (unsigned)n_edges) return;

    int dst = eidx[(size_t)n_edges + e];   // edge_index[1][e]
    if ((unsigned)dst >= (unsigned)batch) return;  // only first `batch` rows matter
    int src = eidx[e];                     // edge_index[0][e]

    float4 v = x4[(size_t)src * 12 + c];
    float* p = out + (size_t)dst * 96 + 48 + (size_t)c * 4;

    asm volatile("global_atomic_add_f32 %0, %1, off"
                 :: "v"(p), "v"(v.x) : "memory");
    asm volatile("global_atomic_add_f32 %0, %1, off offset:4"
                 :: "v"(p), "v"(v.y) : "memory");
    asm volatile("global_atomic_add_f32 %0, %1, off offset:8"
                 :: "v"(p), "v"(v.z) : "memory");
    asm volatile("global_atomic_add_f32 %0, %1, off offset:12"
                 :: "v"(p), "v"(v.w) : "memory");
}

extern "C" void kernel_launch(void* const* d_in, const int* in_sizes, int n_in,
                              void* d_out, int out_size, void* d_ws, size_t ws_size,
                              hipStream_t stream) {
    const float* x    = (const float*)d_in[0];
    const int*   eidx = (const int*)d_in[1];   // edge_index, (2, E) row-major
    (void)d_ws; (void)ws_size; (void)n_in;

    const int n_edges = in_sizes[1] / 2;
    const int batch   = out_size / 96;         // output rows (= batch_size)

    // Phase 1: copy + zero (also initializes the accumulator each call).
    int total_copy = batch * 12;
    int blocks1 = (total_copy + TPB - 1) / TPB;
    copy_zero_async_kernel<<<blocks1, TPB, 0, stream>>>(
        (const float4*)x, (float4*)d_out, total_copy);

    // Phase 2: atomic scatter-add of x[src] into out[dst, 48:96].
    long long total_sc = (long long)n_edges * 12;
    int blocks2 = (int)((total_sc + TPB - 1) / TPB);
    edge_scatter_kernel<<<blocks2, TPB, 0, stream>>>(
        (const float4*)x, eidx, (float*)d_out, n_edges, batch);
}